// GCNEncoder_14860586844771
// MI455X (gfx1250) — compile-verified
//
#include <hip/hip_runtime.h>
#include <hip/hip_bf16.h>

// GCN encoder, 2 layers, D=128.
// Workspace requirement: N*4 (dis) + 2*N*D*4 (H, X1/agg) = ~102.8 MB.

typedef float v2f __attribute__((ext_vector_type(2)));
typedef float v8f __attribute__((ext_vector_type(8)));

#define N_NODES 100000
#define N_EDGES 1600000
#define DIM 128

// ---------------- utility: zero a float buffer ----------------
__global__ void zero_f32(float* __restrict__ p, long n) {
    long i = (long)blockIdx.x * blockDim.x + threadIdx.x;
    if (i < n) p[i] = 0.0f;
}

// ---------------- degree accumulation (in-degree at dst) ----------------
__global__ void deg_accum(const long long* __restrict__ ei,
                          const float* __restrict__ w,
                          float* __restrict__ deg) {
    int e = blockIdx.x * blockDim.x + threadIdx.x;
    if (e < N_EDGES) {
        int dst = (int)ei[(long)N_EDGES + e];
        atomicAdd(&deg[dst], w[e]);
    }
}

// dis[i] = rsqrt(deg[i] + 1)   (self loop weight 1 => deg+1 > 0 always)
__global__ void make_dis(float* __restrict__ deg_dis) {
    int i = blockIdx.x * blockDim.x + threadIdx.x;
    if (i < N_NODES) {
        deg_dis[i] = rsqrtf(deg_dis[i] + 1.0f);
    }
}

// ---------------- H = X @ W via V_WMMA_F32_16X16X4_F32 ----------------
// Block: 256 threads = 8 waves. Block covers rows [16*bid, 16*bid+16),
// wave w covers cols [16*w, 16*w+16). K loop: 32 x WMMA (K=4 each).
//
// Fragment layouts (per CDNA5 ISA 7.12.2, wave32):
//   A (16x4 f32, 2 VGPRs): lanes 0-15 -> M=lane, K={0,1}; lanes 16-31 -> M=lane-16, K={2,3}
//   B (4x16 f32, 2 VGPRs): lanes 0-15 -> N=lane, K={0,1}; lanes 16-31 -> N=lane-16, K={2,3}
//   C/D (16x16 f32, 8 VGPRs): vgpr r: lanes 0-15 -> row r, lanes 16-31 -> row r+8
__launch_bounds__(256)
__global__ void gemm_wmma_f32(const float* __restrict__ X,
                              const float* __restrict__ W,
                              float* __restrict__ H) {
    const int wave = threadIdx.x >> 5;   // 0..7 -> column tile
    const int lane = threadIdx.x & 31;
    const int lid  = lane & 15;
    const int half = lane >> 4;
    const int rowBase = blockIdx.x * 16;
    const int colBase = wave * 16;

    const float* __restrict__ xrow = X + (long)(rowBase + lid) * DIM;

    v8f c = {};
#pragma unroll
    for (int k = 0; k < DIM; k += 4) {
        const int ka = k + 2 * half;
        v2f a;
        a.x = xrow[ka];
        a.y = xrow[ka + 1];
        v2f b;
        b.x = W[(long)ka * DIM + colBase + lid];
        b.y = W[(long)(ka + 1) * DIM + colBase + lid];
        c = __builtin_amdgcn_wmma_f32_16x16x4_f32(
                /*neg_a=*/false, a, /*neg_b=*/false, b,
                /*c_mod=*/(short)0, c, /*reuse_a=*/false, /*reuse_b=*/false);
    }

#pragma unroll
    for (int r = 0; r < 8; ++r) {
        H[(long)(rowBase + r + 8 * half) * DIM + colBase + lid] = c[r];
    }
}

// ---------------- edge scatter: agg[dst] += H[src] * norm ----------------
// One wave per edge; each lane handles 4 consecutive floats (float4 gather,
// 4x global_atomic_add_f32 scatter). H and agg both fit in the 192MB L2.
__launch_bounds__(256)
__global__ void edge_agg(const long long* __restrict__ ei,
                         const float* __restrict__ w,
                         const float* __restrict__ dis,
                         const float* __restrict__ H,
                         float* __restrict__ agg) {
    long t = (long)blockIdx.x * blockDim.x + threadIdx.x;
    long e = t >> 5;
    int lane = (int)(t & 31);
    if (e >= N_EDGES) return;

    int src = (int)ei[e];
    int dst = (int)ei[(long)N_EDGES + e];
    float norm = dis[src] * w[e] * dis[dst];

    const float4* __restrict__ hs = (const float4*)(H + (long)src * DIM);
    float4 v = hs[lane];

    float* __restrict__ ad = agg + (long)dst * DIM + lane * 4;
    atomicAdd(ad + 0, v.x * norm);
    atomicAdd(ad + 1, v.y * norm);
    atomicAdd(ad + 2, v.z * norm);
    atomicAdd(ad + 3, v.w * norm);
}

// ---------------- self-loop + bias + PReLU (in-place capable) ----------------
__global__ void finalize_prelu(const float* __restrict__ agg,
                               const float* __restrict__ H,
                               const float* __restrict__ dis,
                               const float* __restrict__ b,
                               const float* __restrict__ a,
                               float* __restrict__ out) {
    long t = (long)blockIdx.x * blockDim.x + threadIdx.x;
    if (t >= (long)N_NODES * DIM) return;
    int i = (int)(t >> 7);    // node
    int d = (int)(t & 127);   // feature
    float di = dis[i];
    float v = agg[t] + H[t] * (di * di) + b[d];
    out[t] = (v >= 0.0f) ? v : a[d] * v;
}

extern "C" void kernel_launch(void* const* d_in, const int* in_sizes, int n_in,
                              void* d_out, int out_size, void* d_ws, size_t ws_size,
                              hipStream_t stream) {
    const float*      X   = (const float*)d_in[0];
    const long long*  ei  = (const long long*)d_in[1];   // int64 edge_index [2, E]
    const float*      ew  = (const float*)d_in[2];
    const float*      W1  = (const float*)d_in[3];
    const float*      b1  = (const float*)d_in[4];
    const float*      a1  = (const float*)d_in[5];
    const float*      W2  = (const float*)d_in[6];
    const float*      b2  = (const float*)d_in[7];
    const float*      a2  = (const float*)d_in[8];
    float*            out = (float*)d_out;

    const long ND = (long)N_NODES * DIM;

    // Workspace layout (16B aligned: N_NODES*4 = 400000 bytes, /16 = 25000).
    float* dis = (float*)d_ws;           // N floats: deg, then dis (in place)
    float* H   = dis + N_NODES;          // N*D floats: GEMM output (both layers)
    float* X1  = H + ND;                 // N*D floats: agg1, then layer-1 output (in place)

    const int B = 256;
    const int gN    = (N_NODES + B - 1) / B;
    const int gND   = (int)((ND + B - 1) / B);
    const int gE    = (N_EDGES + B - 1) / B;
    const int gE32  = (int)(((long)N_EDGES * 32 + B - 1) / B);
    const int gGemm = N_NODES / 16;      // 6250 (exact)

    // ---- normalization ----
    zero_f32<<<gN, B, 0, stream>>>(dis, N_NODES);
    deg_accum<<<gE, B, 0, stream>>>(ei, ew, dis);
    make_dis<<<gN, B, 0, stream>>>(dis);

    // ---- layer 1 ----
    zero_f32<<<gND, B, 0, stream>>>(X1, ND);                       // agg1 = 0
    gemm_wmma_f32<<<gGemm, B, 0, stream>>>(X, W1, H);              // H = X @ W1
    edge_agg<<<gE32, B, 0, stream>>>(ei, ew, dis, H, X1);          // agg1 += scatter
    finalize_prelu<<<gND, B, 0, stream>>>(X1, H, dis, b1, a1, X1); // X1 = prelu(...)

    // ---- layer 2 ----
    zero_f32<<<gND, B, 0, stream>>>(out, ND);                      // agg2 = 0 (d_out)
    gemm_wmma_f32<<<gGemm, B, 0, stream>>>(X1, W2, H);             // H = X1 @ W2
    edge_agg<<<gE32, B, 0, stream>>>(ei, ew, dis, H, out);
    finalize_prelu<<<gND, B, 0, stream>>>(out, H, dis, b2, a2, out);
}